// GATBackbone_28879360098775
// MI455X (gfx1250) — compile-verified
//
#include <hip/hip_runtime.h>
#include <cstdint>
#include <cstddef>

#define N_NODES 50000
#define N_EDGES 800000
#define N_GRAPHS 64
#define FDIM 128
#define NHEAD 4
#define CDIM 32
#define NEG_SLOPE 0.2f
#define EPS_F 1e-16f

typedef __attribute__((ext_vector_type(2))) float v2f;
typedef __attribute__((ext_vector_type(8))) float v8f;

// ---------------------------------------------------------------------------
// GEMM: xl = h @ W^T + b   (h: [N,128], W: [128,128] row-major [out,in])
// One wave computes a 16x128 strip via V_WMMA_F32_16X16X4_F32.
// LDS holds W as [col][k] with an even-XOR swizzle on k:
//   Wlds[col*128 + (k ^ ((2*col)&126))]
// -> (k,k+1) pairs stay contiguous (single ds_load_b64 per B operand) and both
//    the staging stores and reads are bank-conflict-free (XOR bijection mod 64).
// ---------------------------------------------------------------------------
__global__ __launch_bounds__(128)
void gat_gemm_wmma(const float* __restrict__ h, const float* __restrict__ Wg,
                   const float* __restrict__ bg, float* __restrict__ xl,
                   int ntiles) {
  __shared__ float Wlds[FDIM * FDIM];
  const int tid = threadIdx.x;
  for (int i = tid; i < FDIM * FDIM; i += 128) {
    const int r = i >> 7;    // output column (row of W)
    const int c = i & 127;   // k (column of W)
    Wlds[r * FDIM + (c ^ ((2 * r) & 126))] = Wg[i];
  }
  __syncthreads();

  const int wave = tid >> 5;
  const int lane = tid & 31;
  const int m    = lane & 15;   // A row / B,C,D column within tile
  const int kh   = lane >> 4;   // K-half selector
  const int tile = blockIdx.x * 4 + wave;   // wave-uniform guard -> EXEC all 1s
  if (tile >= ntiles) return;
  const int row0 = tile * 16;

  const float* hrow = h + (size_t)(row0 + m) * FDIM;

  float bias[8];
#pragma unroll
  for (int j = 0; j < 8; ++j) bias[j] = bg[j * 16 + m];

  const v8f vzero = {0.f, 0.f, 0.f, 0.f, 0.f, 0.f, 0.f, 0.f};
  v8f acc[8];
#pragma unroll
  for (int j = 0; j < 8; ++j) acc[j] = vzero;

  for (int kb = 0; kb < FDIM / 4; ++kb) {
    const int kk = kb * 4 + 2 * kh;                 // even -> 8B aligned
    v2f a = *(const v2f*)(hrow + kk);               // A[m][kk], A[m][kk+1]
#pragma unroll
    for (int j = 0; j < 8; ++j) {
      const int col = j * 16 + m;
      // b = { W[col][kk], W[col][kk+1] } = { B[kk][col], B[kk+1][col] }
      v2f b = *(const v2f*)&Wlds[col * FDIM + (kk ^ ((2 * col) & 126))];
      acc[j] = __builtin_amdgcn_wmma_f32_16x16x4_f32(
          false, a, false, b, (short)0, acc[j], false, false);
    }
  }

#pragma unroll
  for (int j = 0; j < 8; ++j) {
    const int col = j * 16 + m;
#pragma unroll
    for (int v = 0; v < 8; ++v) {
      const int mr = v + 8 * kh;   // C/D layout: VGPR v -> row v (lanes 0-15), v+8 (16-31)
      xl[(size_t)(row0 + mr) * FDIM + col] = acc[j][v] + bias[j];
    }
  }
}

// ---------------------------------------------------------------------------
// Per-node attention coefficients: alpha_{l,r}[n,h] = sum_c xl[n,h,c]*att[h,c]
// One wave (32 lanes) per (node, head); wave32 shuffle reduction.
// ---------------------------------------------------------------------------
__global__ __launch_bounds__(128)
void gat_alpha(const float* __restrict__ xl, const float* __restrict__ attl,
               const float* __restrict__ attr, float* __restrict__ al,
               float* __restrict__ ar) {
  const int n  = blockIdx.x;
  const int hh = threadIdx.x >> 5;
  const int c  = threadIdx.x & 31;
  const float v = xl[(size_t)n * FDIM + hh * CDIM + c];
  float sl = v * attl[hh * CDIM + c];
  float sr = v * attr[hh * CDIM + c];
#pragma unroll
  for (int off = 16; off > 0; off >>= 1) {
    sl += __shfl_xor(sl, off, 32);
    sr += __shfl_xor(sr, off, 32);
  }
  if (c == 0) {
    al[n * NHEAD + hh] = sl;
    ar[n * NHEAD + hh] = sr;
  }
}

// ---------------------------------------------------------------------------
// Per-layer init: zero output accumulator (b128), segmax = ordered(-inf), denom = 0
// ---------------------------------------------------------------------------
__global__ void gat_init(float4* __restrict__ out4, int* __restrict__ segmax,
                         float* __restrict__ denom) {
  const int idx = blockIdx.x * blockDim.x + threadIdx.x;
  if (idx < N_NODES * (FDIM / 4)) out4[idx] = make_float4(0.f, 0.f, 0.f, 0.f);
  if (idx < N_NODES * NHEAD) {
    segmax[idx] = (int)0x80000000;  // ordered-int -inf
    denom[idx] = 0.f;
  }
}

__device__ __forceinline__ int f2ord(float f) {
  const int b = __float_as_int(f);
  return b >= 0 ? b : (b ^ 0x7fffffff);
}
__device__ __forceinline__ float ord2f(int k) {
  return __int_as_float(k >= 0 ? k : (k ^ 0x7fffffff));
}

// Pass 1: e = leaky_relu(alpha_r[dst] + alpha_l[src]); segment max via int atomics
__global__ void gat_edge_logits(const int* __restrict__ ei,
                                const float* __restrict__ al,
                                const float* __restrict__ ar,
                                float* __restrict__ ebuf,
                                int* __restrict__ segmax) {
  const int idx = blockIdx.x * blockDim.x + threadIdx.x;
  if (idx >= N_EDGES * NHEAD) return;
  const int e = idx >> 2, hh = idx & 3;
  const int s = ei[e];
  const int d = ei[N_EDGES + e];
  float v = ar[d * NHEAD + hh] + al[s * NHEAD + hh];
  v = v > 0.f ? v : NEG_SLOPE * v;
  ebuf[idx] = v;
  atomicMax(&segmax[d * NHEAD + hh], f2ord(v));
}

// Pass 2: e <- exp(e - max); denom[dst] += e
__global__ void gat_edge_exp(const int* __restrict__ ei,
                             float* __restrict__ ebuf,
                             const int* __restrict__ segmax,
                             float* __restrict__ denom) {
  const int idx = blockIdx.x * blockDim.x + threadIdx.x;
  if (idx >= N_EDGES * NHEAD) return;
  const int e = idx >> 2, hh = idx & 3;
  const int d = ei[N_EDGES + e];
  const float mx = ord2f(segmax[d * NHEAD + hh]);
  const float e2 = __expf(ebuf[idx] - mx);
  ebuf[idx] = e2;
  atomicAdd(&denom[d * NHEAD + hh], e2);
}

// Pass 3: out[dst] += xl[src] * (e / (denom[dst] + eps))
// 32 lanes per edge, float4 (b128) gathers; atomics into the L2-resident out.
__global__ __launch_bounds__(128)
void gat_edge_msg(const int* __restrict__ ei, const float* __restrict__ xl,
                  const float* __restrict__ ebuf,
                  const float* __restrict__ denom, float* __restrict__ out) {
  const int e    = blockIdx.x * 4 + (threadIdx.x >> 5);
  const int lane = threadIdx.x & 31;
  const int hh   = lane >> 3;              // head of features [lane*4, lane*4+3]
  const int s = ei[e];
  const int d = ei[N_EDGES + e];
  const float a = ebuf[e * NHEAD + hh] / (denom[d * NHEAD + hh] + EPS_F);
  const float4 v = *(const float4*)(xl + (size_t)s * FDIM + lane * 4);
  float* o = out + (size_t)d * FDIM + lane * 4;
  atomicAdd(o + 0, v.x * a);
  atomicAdd(o + 1, v.y * a);
  atomicAdd(o + 2, v.z * a);
  atomicAdd(o + 3, v.w * a);
}

__global__ void gat_relu(float4* __restrict__ x, int n4) {
  const int i = blockIdx.x * blockDim.x + threadIdx.x;
  if (i < n4) {
    float4 v = x[i];
    v.x = fmaxf(v.x, 0.f);
    v.y = fmaxf(v.y, 0.f);
    v.z = fmaxf(v.z, 0.f);
    v.w = fmaxf(v.w, 0.f);
    x[i] = v;
  }
}

__global__ void gat_zero(float* __restrict__ x, int n) {
  const int i = blockIdx.x * blockDim.x + threadIdx.x;
  if (i < n) x[i] = 0.f;
}

__global__ void gat_pool(const float* __restrict__ h, const int* __restrict__ batch,
                         float* __restrict__ pooled) {
  const int idx = blockIdx.x * blockDim.x + threadIdx.x;  // over N_NODES * 32
  if (idx >= N_NODES * (FDIM / 4)) return;
  const int n = idx >> 5, q = (idx & 31) * 4;
  const float4 v = *(const float4*)(h + (size_t)n * FDIM + q);
  float* o = pooled + batch[n] * FDIM + q;
  atomicAdd(o + 0, v.x);
  atomicAdd(o + 1, v.y);
  atomicAdd(o + 2, v.z);
  atomicAdd(o + 3, v.w);
}

__global__ void gat_proj(const float* __restrict__ pooled,
                         const float* __restrict__ pW,
                         const float* __restrict__ pb, float* __restrict__ out) {
  const int idx = blockIdx.x * blockDim.x + threadIdx.x;
  if (idx >= N_GRAPHS * 32) return;
  const int g = idx >> 5, o = idx & 31;
  float s = pb[o];
#pragma unroll 4
  for (int k = 0; k < FDIM; ++k) s += pooled[g * FDIM + k] * pW[o * FDIM + k];
  out[idx] = s;
}

// ---------------------------------------------------------------------------
extern "C" void kernel_launch(void* const* d_in, const int* in_sizes, int n_in,
                              void* d_out, int out_size, void* d_ws, size_t ws_size,
                              hipStream_t stream) {
  (void)in_sizes; (void)n_in; (void)out_size; (void)ws_size;
  const float* x     = (const float*)d_in[0];
  const int*   ei    = (const int*)  d_in[1];
  const int*   batch = (const int*)  d_in[2];
  const float* Ws    = (const float*)d_in[3];
  const float* bs    = (const float*)d_in[4];
  const float* attls = (const float*)d_in[5];
  const float* attrs = (const float*)d_in[6];
  const float* pW    = (const float*)d_in[7];
  const float* pb    = (const float*)d_in[8];
  float* out = (float*)d_out;

  uintptr_t p = (uintptr_t)d_ws;
  auto carve = [&](size_t bytes) -> void* {
    uintptr_t q = (p + 255) & ~(uintptr_t)255;
    p = q + bytes;
    return (void*)q;
  };
  float* bufA   = (float*)carve((size_t)N_NODES * FDIM * sizeof(float));   // xl
  float* bufB   = (float*)carve((size_t)N_NODES * FDIM * sizeof(float));   // h / out
  float* al     = (float*)carve((size_t)N_NODES * NHEAD * sizeof(float));
  float* ar     = (float*)carve((size_t)N_NODES * NHEAD * sizeof(float));
  int*   segmax = (int*)  carve((size_t)N_NODES * NHEAD * sizeof(int));
  float* denom  = (float*)carve((size_t)N_NODES * NHEAD * sizeof(float));
  float* ebuf   = (float*)carve((size_t)N_EDGES * NHEAD * sizeof(float));
  float* pooled = (float*)carve((size_t)N_GRAPHS * FDIM * sizeof(float));

  const int ntiles = N_NODES / 16;              // 3125 (exact)
  const int gemm_blocks = (ntiles + 3) / 4;     // 782 blocks x 4 waves
  const int nodef4 = N_NODES * (FDIM / 4);      // 1.6M float4s
  const int nodef4_blocks = (nodef4 + 255) / 256;
  const int edgeh_blocks = (N_EDGES * NHEAD + 255) / 256;

  const float* hcur = x;
  for (int l = 0; l < 3; ++l) {
    gat_gemm_wmma<<<gemm_blocks, 128, 0, stream>>>(
        hcur, Ws + (size_t)l * FDIM * FDIM, bs + (size_t)l * FDIM, bufA, ntiles);
    gat_alpha<<<N_NODES, 128, 0, stream>>>(
        bufA, attls + (size_t)l * NHEAD * CDIM, attrs + (size_t)l * NHEAD * CDIM, al, ar);
    gat_init<<<nodef4_blocks, 256, 0, stream>>>((float4*)bufB, segmax, denom);
    gat_edge_logits<<<edgeh_blocks, 256, 0, stream>>>(ei, al, ar, ebuf, segmax);
    gat_edge_exp<<<edgeh_blocks, 256, 0, stream>>>(ei, ebuf, segmax, denom);
    gat_edge_msg<<<N_EDGES / 4, 128, 0, stream>>>(ei, bufA, ebuf, denom, bufB);
    gat_relu<<<nodef4_blocks, 256, 0, stream>>>((float4*)bufB, nodef4);
    hcur = bufB;   // next layer reads bufB; its GEMM finishes before gat_init rezeroes it
  }

  gat_zero<<<(N_GRAPHS * FDIM + 255) / 256, 256, 0, stream>>>(pooled, N_GRAPHS * FDIM);
  gat_pool<<<nodef4_blocks, 256, 0, stream>>>(bufB, batch, pooled);
  gat_proj<<<(N_GRAPHS * 32 + 255) / 256, 256, 0, stream>>>(pooled, pW, pb, out);
}